// HeteroGNN_27633819582984
// MI455X (gfx1250) — compile-verified
//
#include <hip/hip_runtime.h>

// ---------------- problem constants (from reference) ----------------
#define NSUB    8000
#define NREG    40000
#define DIN     128
#define HD      64        // hidden width H
#define NHEADS  2
#define ODIM    3
#define ESR     800000
#define ERR     640000
#define NLAYERS 2
#define EPSV    1e-5f
#define LSLOPE  0.2f

#define GEMM_WAVES 4      // waves per GEMM block
#define MT 2              // 16-row M tiles per wave (wave computes 32x64 of C)

typedef __attribute__((ext_vector_type(2))) float v2f;
typedef __attribute__((ext_vector_type(8))) float v8f;

// ---------------- CDNA5 f32 WMMA: D = A(16x4) * B(4x16) + C ----------------
__device__ inline v8f wmma_f32_16x16x4(v2f a, v2f b, v8f c) {
  // 8 args: (neg_a, A, neg_b, B, c_mod, C, reuse_a, reuse_b)
  return __builtin_amdgcn_wmma_f32_16x16x4_f32(false, a, false, b, (short)0, c,
                                               false, false);
}

// monotonic float<->uint encoding for atomicMax-based segment max
__device__ inline unsigned fenc(float f) {
  unsigned u = __float_as_uint(f);
  return (u & 0x80000000u) ? ~u : (u | 0x80000000u);
}
__device__ inline float fdec(unsigned u) {
  return (u & 0x80000000u) ? __uint_as_float(u & 0x7FFFFFFFu)
                           : __uint_as_float(~u);
}

// edge (src,dst) with appended self loops for e >= Err
__device__ inline void edge_sd(const int* __restrict__ src,
                               const int* __restrict__ dst, int e, int Err,
                               int& s, int& d) {
  if (e < Err) { s = src[e]; d = dst[e]; }
  else         { s = e - Err; d = s; }
}

// ---------------- segment_sum of gathered rows: out[dst] += X[src] ----------------
__global__ void k_scatter_add(const int* __restrict__ src,
                              const int* __restrict__ dst,
                              const float* __restrict__ X,
                              float* __restrict__ out, int E, int d) {
  int t = blockIdx.x * blockDim.x + threadIdx.x;
  int chunks = d >> 2;                       // float4 chunks per row
  if (t >= E * chunks) return;
  int e = t / chunks;
  int c = (t - e * chunks) << 2;
  int s = src[e], dd = dst[e];
  float4 v = *(const float4*)(X + (size_t)s * d + c);
  float* o = out + (size_t)dd * d + c;
  atomicAdd(o + 0, v.x); atomicAdd(o + 1, v.y);
  atomicAdd(o + 2, v.z); atomicAdd(o + 3, v.w);
}

// ---------------- WMMA GEMM with LDS-staged, k-paired weights ----------------
// C[32 x 64 per wave] = A1@W1 (+ A2@W2) (+ bias). A row-major packed (lda == K),
// W row-major (leading dim ldw); the 64-column slice at ncol0 is staged in LDS
// as float2 pairs over K: sw[q][col] = {W[2q][col], W[2q+1][col]} so every
// WMMA B fragment is one aligned ds_load_b64. M must be a multiple of 32.
__global__ void k_gemm64(const float* __restrict__ A1, const float* __restrict__ W1,
                         int K1,
                         const float* __restrict__ A2, const float* __restrict__ W2,
                         int K2,
                         const float* __restrict__ bias, float* __restrict__ C,
                         int M, int ldw, int ldc, int ncol0) {
  // worst case: two (128/2)x64 float2 slabs = 64 KB (WGP has 320 KB LDS)
  __shared__ v2f sw[2][(DIN / 2) * HD];

  int tid = threadIdx.y * 32 + threadIdx.x;
  for (int i = tid; i < (K1 >> 1) * HD; i += GEMM_WAVES * 32) {
    int q = i >> 6, c = i & 63;
    v2f w;
    w.x = W1[(size_t)(2 * q) * ldw + ncol0 + c];
    w.y = W1[(size_t)(2 * q + 1) * ldw + ncol0 + c];
    sw[0][i] = w;
  }
  if (A2) {
    for (int i = tid; i < (K2 >> 1) * HD; i += GEMM_WAVES * 32) {
      int q = i >> 6, c = i & 63;
      v2f w;
      w.x = W2[(size_t)(2 * q) * ldw + ncol0 + c];
      w.y = W2[(size_t)(2 * q + 1) * ldw + ncol0 + c];
      sw[1][i] = w;
    }
  }
  __syncthreads();

  int grp = blockIdx.x * GEMM_WAVES + threadIdx.y;  // group of MT row-tiles
  if (grp * (16 * MT) >= M) return;                 // wave-uniform: EXEC all-1
  int lane = threadIdx.x;                           // 0..31 (wave32)
  int lm = lane & 15;                               // A row in tile / B column
  int kh = lane >> 4;                               // K pair-offset 0 or 1
  int kq = kh << 1;                                 // K sub-offset 0 or 2

  v8f acc[MT][4];
#pragma unroll
  for (int mt = 0; mt < MT; ++mt)
#pragma unroll
    for (int t = 0; t < 4; ++t) acc[mt][t] = 0.f;

  const float* arow[MT];
#pragma unroll
  for (int mt = 0; mt < MT; ++mt)
    arow[mt] = A1 + ((size_t)(grp * MT + mt) * 16 + lm) * K1;

#pragma unroll 4
  for (int k = 0; k < K1; k += 4) {
    v2f a[MT];
#pragma unroll
    for (int mt = 0; mt < MT; ++mt) {
      a[mt].x = arow[mt][k + kq];
      a[mt].y = arow[mt][k + kq + 1];
    }
    int qb = ((k >> 1) + kh) * HD + lm;             // paired-K LDS row
#pragma unroll
    for (int t = 0; t < 4; ++t) {
      v2f b = sw[0][qb + t * 16];                   // one ds_load_b64
#pragma unroll
      for (int mt = 0; mt < MT; ++mt)
        acc[mt][t] = wmma_f32_16x16x4(a[mt], b, acc[mt][t]);
    }
  }
  if (A2) {
    const float* a2row[MT];
#pragma unroll
    for (int mt = 0; mt < MT; ++mt)
      a2row[mt] = A2 + ((size_t)(grp * MT + mt) * 16 + lm) * K2;
#pragma unroll 4
    for (int k = 0; k < K2; k += 4) {
      v2f a[MT];
#pragma unroll
      for (int mt = 0; mt < MT; ++mt) {
        a[mt].x = a2row[mt][k + kq];
        a[mt].y = a2row[mt][k + kq + 1];
      }
      int qb = ((k >> 1) + kh) * HD + lm;
#pragma unroll
      for (int t = 0; t < 4; ++t) {
        v2f b = sw[1][qb + t * 16];
#pragma unroll
        for (int mt = 0; mt < MT; ++mt)
          acc[mt][t] = wmma_f32_16x16x4(a[mt], b, acc[mt][t]);
      }
    }
  }

  // C/D layout: VGPR r, lanes 0-15 -> M=r, lanes 16-31 -> M=8+r; N = lane&15
  int rm = kh << 3;
#pragma unroll
  for (int mt = 0; mt < MT; ++mt) {
#pragma unroll
    for (int r = 0; r < 8; ++r) {
      int row = (grp * MT + mt) * 16 + rm + r;
#pragma unroll
      for (int t = 0; t < 4; ++t) {
        int col = ncol0 + t * 16 + lm;
        float v = acc[mt][t][r];
        if (bias) v += bias[col];
        C[(size_t)row * ldc + col] = v;
      }
    }
  }
}

// ---------------- GAT attention scores: a_s/a_d[node,head] ----------------
__global__ void k_gat_scores(const float* __restrict__ h,
                             const float* __restrict__ asrc,
                             const float* __restrict__ adst,
                             float* __restrict__ a_s, float* __restrict__ a_d) {
  int t = blockIdx.x * blockDim.x + threadIdx.x;
  if (t >= NREG * NHEADS) return;
  int head = t & 1, node = t >> 1;
  const float* hp = h + (size_t)node * (NHEADS * HD) + head * HD;
  const float* ps = asrc + head * HD;
  const float* pd = adst + head * HD;
  float ss = 0.f, sd = 0.f;
  for (int f = 0; f < HD; ++f) { float v = hp[f]; ss += v * ps[f]; sd += v * pd[f]; }
  a_s[t] = ss; a_d[t] = sd;
}

__global__ void k_gat_max(const int* __restrict__ src, const int* __restrict__ dst,
                          int Err, int total, const float* __restrict__ a_s,
                          const float* __restrict__ a_d, unsigned* __restrict__ emax) {
  int t = blockIdx.x * blockDim.x + threadIdx.x;
  if (t >= total * NHEADS) return;
  int head = t & 1, e = t >> 1, s, d;
  edge_sd(src, dst, e, Err, s, d);
  float v = a_s[s * 2 + head] + a_d[d * 2 + head];
  v = fmaxf(v, LSLOPE * v);                  // leaky relu
  atomicMax(&emax[d * 2 + head], fenc(v));
}

__global__ void k_gat_sum(const int* __restrict__ src, const int* __restrict__ dst,
                          int Err, int total, const float* __restrict__ a_s,
                          const float* __restrict__ a_d,
                          const unsigned* __restrict__ emax,
                          float* __restrict__ esum) {
  int t = blockIdx.x * blockDim.x + threadIdx.x;
  if (t >= total * NHEADS) return;
  int head = t & 1, e = t >> 1, s, d;
  edge_sd(src, dst, e, Err, s, d);
  int id = d * 2 + head;
  float v = a_s[s * 2 + head] + a_d[id];
  v = fmaxf(v, LSLOPE * v);
  atomicAdd(&esum[id], __expf(v - fdec(emax[id])));
}

// weighted aggregation; 0.5 (head mean) folded into the weight so both heads
// accumulate into one NREG x 64 buffer. One thread = (edge, head, 16-float chunk).
__global__ void k_gat_agg(const int* __restrict__ src, const int* __restrict__ dst,
                          int Err, int total, const float* __restrict__ a_s,
                          const float* __restrict__ a_d,
                          const unsigned* __restrict__ emax,
                          const float* __restrict__ esum,
                          const float* __restrict__ h, float* __restrict__ outg) {
  int t = blockIdx.x * blockDim.x + threadIdx.x;
  if (t >= total * NHEADS * 4) return;
  int chunk = t & 3, head = (t >> 2) & 1, e = t >> 3, s, d;
  edge_sd(src, dst, e, Err, s, d);
  int id = d * 2 + head;
  float v = a_s[s * 2 + head] + a_d[id];
  v = fmaxf(v, LSLOPE * v);
  float w = 0.5f * __expf(v - fdec(emax[id])) / esum[id];
  const float* hp = h + (size_t)s * (NHEADS * HD) + head * HD + chunk * 16;
  float* op = outg + (size_t)d * HD + chunk * 16;
#pragma unroll
  for (int i = 0; i < 16; i += 4) {
    float4 hv = *(const float4*)(hp + i);
    atomicAdd(op + i + 0, w * hv.x); atomicAdd(op + i + 1, w * hv.y);
    atomicAdd(op + i + 2, w * hv.z); atomicAdd(op + i + 3, w * hv.w);
  }
}

// ---------------- BN(eval) + ReLU epilogue (optionally adds GAT agg + gat_b) ----------------
__global__ void k_bn_relu(const float* __restrict__ in, const float* __restrict__ addg,
                          const float* __restrict__ gat_b,
                          const float* __restrict__ bn_g, const float* __restrict__ bn_b,
                          const float* __restrict__ bn_m, const float* __restrict__ bn_v,
                          float* __restrict__ outp, int n) {
  int t = blockIdx.x * blockDim.x + threadIdx.x;
  if (t >= n) return;
  int f = t & (HD - 1);
  float v = in[t];
  if (addg) v += addg[t] + gat_b[f];
  v = (v - bn_m[f]) * rsqrtf(bn_v[f] + EPSV) * bn_g[f] + bn_b[f];
  outp[t] = fmaxf(v, 0.f);
}

// ---------------- final linear 64 -> 3 ----------------
__global__ void k_final(const float* __restrict__ X, const float* __restrict__ W,
                        const float* __restrict__ b, float* __restrict__ out, int M) {
  int r = blockIdx.x * blockDim.x + threadIdx.x;
  if (r >= M) return;
  const float* x = X + (size_t)r * HD;
  float a0 = b[0], a1 = b[1], a2 = b[2];
  for (int f = 0; f < HD; ++f) {
    float v = x[f];
    a0 += v * W[f * 3 + 0]; a1 += v * W[f * 3 + 1]; a2 += v * W[f * 3 + 2];
  }
  out[r * 3 + 0] = a0; out[r * 3 + 1] = a1; out[r * 3 + 2] = a2;
}

// ---------------- host orchestration ----------------
static inline int cdiv(long long a, long long b) { return (int)((a + b - 1) / b); }

extern "C" void kernel_launch(void* const* d_in, const int* in_sizes, int n_in,
                              void* d_out, int out_size, void* d_ws, size_t ws_size,
                              hipStream_t stream) {
  (void)in_sizes; (void)n_in; (void)out_size; (void)ws_size;

  const float* x_subject = (const float*)d_in[0];
  const float* x_region  = (const float*)d_in[1];
  const int* sr_src = (const int*)d_in[2];
  const int* sr_dst = (const int*)d_in[3];
  const int* rs_src = (const int*)d_in[4];
  const int* rs_dst = (const int*)d_in[5];
  const int* rr_src = (const int*)d_in[6];
  const int* rr_dst = (const int*)d_in[7];

  // params flattened per jax pytree order (alphabetical keys), 14 per layer:
  // 0 bn_b 1 bn_g 2 bn_m 3 bn_v 4 gat_W 5 gat_adst 6 gat_asrc 7 gat_b
  // 8 rs_Wrel 9 rs_Wroot 10 rs_b 11 sr_Wrel 12 sr_Wroot 13 sr_b
  const int PBASE = 8;
  const int LBASE = PBASE + NLAYERS * 14;    // lin_params: 0 Wr 1 Ws 2 br 3 bs

  // ---- workspace carve (floats) ----
  float* p = (float*)d_ws;
  float* agg_r = p; p += (size_t)NREG * DIN;       // segment_sum subj->region
  float* agg_s = p; p += (size_t)NSUB * DIN;       // segment_sum region->subj
  float* hbuf  = p; p += (size_t)NREG * (NHEADS * HD);
  float* a_s   = p; p += (size_t)NREG * NHEADS;
  float* a_d   = p; p += (size_t)NREG * NHEADS;
  unsigned* emax = (unsigned*)p; p += (size_t)NREG * NHEADS;
  float* esum  = p; p += (size_t)NREG * NHEADS;
  float* outg  = p; p += (size_t)NREG * HD;        // head-averaged GAT agg
  float* out_r = p; p += (size_t)NREG * HD;
  float* out_s = p; p += (size_t)NSUB * HD;
  float* xs_buf[2]; float* xr_buf[2];
  xs_buf[0] = p; p += (size_t)NSUB * HD;
  xr_buf[0] = p; p += (size_t)NREG * HD;
  xs_buf[1] = p; p += (size_t)NSUB * HD;
  xr_buf[1] = p; p += (size_t)NREG * HD;

  const float* xs_cur = x_subject;
  const float* xr_cur = x_region;
  int d = DIN;
  const int totalE = ERR + NREG;                   // rr edges + self loops

  const dim3 gblk(32, GEMM_WAVES);
  const int g_reg = cdiv(NREG / (16 * MT), GEMM_WAVES);
  const int g_sub = cdiv(NSUB / (16 * MT), GEMM_WAVES);

  for (int l = 0; l < NLAYERS; ++l) {
    const int pb = PBASE + l * 14;
    const float* bn_b  = (const float*)d_in[pb + 0];
    const float* bn_g  = (const float*)d_in[pb + 1];
    const float* bn_m  = (const float*)d_in[pb + 2];
    const float* bn_v  = (const float*)d_in[pb + 3];
    const float* gat_W = (const float*)d_in[pb + 4];
    const float* gat_ad= (const float*)d_in[pb + 5];
    const float* gat_as= (const float*)d_in[pb + 6];
    const float* gat_b = (const float*)d_in[pb + 7];
    const float* rs_Wrel = (const float*)d_in[pb + 8];
    const float* rs_Wroot= (const float*)d_in[pb + 9];
    const float* rs_b    = (const float*)d_in[pb + 10];
    const float* sr_Wrel = (const float*)d_in[pb + 11];
    const float* sr_Wroot= (const float*)d_in[pb + 12];
    const float* sr_b    = (const float*)d_in[pb + 13];

    // zero accumulators (graph-capturable memset nodes)
    hipMemsetAsync(agg_r, 0, (size_t)NREG * d * sizeof(float), stream);
    hipMemsetAsync(agg_s, 0, (size_t)NSUB * d * sizeof(float), stream);
    hipMemsetAsync(emax, 0, (size_t)NREG * NHEADS * sizeof(unsigned), stream);
    hipMemsetAsync(esum, 0, (size_t)NREG * NHEADS * sizeof(float), stream);
    hipMemsetAsync(outg, 0, (size_t)NREG * HD * sizeof(float), stream);

    // edge aggregation (L2-resident gathers + f32 atomics)
    {
      int n = ESR * (d >> 2);
      k_scatter_add<<<cdiv(n, 256), 256, 0, stream>>>(sr_src, sr_dst, xs_cur, agg_r, ESR, d);
      k_scatter_add<<<cdiv(n, 256), 256, 0, stream>>>(rs_src, rs_dst, xr_cur, agg_s, ESR, d);
    }

    // dense transforms via f32 WMMA (weights staged in LDS, k-paired)
    // out_r = agg_r @ sr_Wrel + xr @ sr_Wroot + sr_b
    k_gemm64<<<g_reg, gblk, 0, stream>>>(
        agg_r, sr_Wrel, d, xr_cur, sr_Wroot, d, sr_b, out_r, NREG, HD, HD, 0);
    // out_s = agg_s @ rs_Wrel + xs @ rs_Wroot + rs_b
    k_gemm64<<<g_sub, gblk, 0, stream>>>(
        agg_s, rs_Wrel, d, xs_cur, rs_Wroot, d, rs_b, out_s, NSUB, HD, HD, 0);
    // h = xr @ gat_W   (128 cols -> two 64-col halves)
    k_gemm64<<<g_reg, gblk, 0, stream>>>(
        xr_cur, gat_W, d, nullptr, nullptr, 0, nullptr, hbuf, NREG, 128, 128, 0);
    k_gemm64<<<g_reg, gblk, 0, stream>>>(
        xr_cur, gat_W, d, nullptr, nullptr, 0, nullptr, hbuf, NREG, 128, 128, 64);

    // attention scores, softmax over incoming edges, weighted aggregation
    k_gat_scores<<<cdiv(NREG * NHEADS, 256), 256, 0, stream>>>(hbuf, gat_as, gat_ad, a_s, a_d);
    k_gat_max<<<cdiv(totalE * NHEADS, 256), 256, 0, stream>>>(
        rr_src, rr_dst, ERR, totalE, a_s, a_d, emax);
    k_gat_sum<<<cdiv(totalE * NHEADS, 256), 256, 0, stream>>>(
        rr_src, rr_dst, ERR, totalE, a_s, a_d, emax, esum);
    k_gat_agg<<<cdiv(totalE * NHEADS * 4, 256), 256, 0, stream>>>(
        rr_src, rr_dst, ERR, totalE, a_s, a_d, emax, esum, hbuf, outg);

    // epilogues: region adds GAT output + gat_b, both get shared BN + ReLU
    k_bn_relu<<<cdiv(NREG * HD, 256), 256, 0, stream>>>(
        out_r, outg, gat_b, bn_g, bn_b, bn_m, bn_v, xr_buf[l], NREG * HD);
    k_bn_relu<<<cdiv(NSUB * HD, 256), 256, 0, stream>>>(
        out_s, nullptr, nullptr, bn_g, bn_b, bn_m, bn_v, xs_buf[l], NSUB * HD);

    xs_cur = xs_buf[l];
    xr_cur = xr_buf[l];
    d = HD;
  }

  // final linears: (xs @ Ws + bs, xr @ Wr + br), concatenated subject-first
  const float* Wr = (const float*)d_in[LBASE + 0];
  const float* Ws = (const float*)d_in[LBASE + 1];
  const float* br = (const float*)d_in[LBASE + 2];
  const float* bs = (const float*)d_in[LBASE + 3];
  float* out_sub = (float*)d_out;
  float* out_reg = out_sub + (size_t)NSUB * ODIM;
  k_final<<<cdiv(NSUB, 256), 256, 0, stream>>>(xs_cur, Ws, bs, out_sub, NSUB);
  k_final<<<cdiv(NREG, 256), 256, 0, stream>>>(xr_cur, Wr, br, out_reg, NREG);
}